// Decoder_29695403885316
// MI455X (gfx1250) — compile-verified
//
#include <hip/hip_runtime.h>
#include <math.h>

// ---- problem constants (from reference) ------------------------------------
#define VOCAB 32000
#define EMB   256
#define HID   512
#define SRC   128
#define BATCH 64
#define STEPS 51
#define SOS   1
#define ROWS  (BATCH * STEPS)   // 3264 = 204 * 16
#define MT16  (ROWS / 16)       // 204

typedef __attribute__((ext_vector_type(2))) float v2f;
typedef __attribute__((ext_vector_type(4))) float v4f;
typedef __attribute__((ext_vector_type(8))) float v8f;

// CDNA5 fp32 WMMA: D[16x16] += A[16x4] * B[4x16], full fp32 precision.
// A frag (per lane): A[l16][k0+2*half + {0,1}]   (l16=lane&15, half=lane>>4)
// B frag (per lane): B[k0+2*half + {0,1}][l16]
// C/D:  c[g] = D[g + 8*half][l16]
__device__ __forceinline__ v8f wmma4(v2f a, v2f b, v8f c) {
  return __builtin_amdgcn_wmma_f32_16x16x4_f32(
      false, a, false, b, (short)0, c, false, false);
}

// ============================================================================
// Kernel 1: embedding gather + gi = x*w_ih^T (WMMA) + GRU gate math -> h
// ============================================================================
__global__ __launch_bounds__(256) void k1_gru(
    const int* __restrict__ target, const float* __restrict__ embedding,
    const float* __restrict__ w_ih, const float* __restrict__ b_ih,
    const float* __restrict__ b_hh, float* __restrict__ hbuf,
    float* __restrict__ dhid) {
  __shared__ float xA[16 * EMB];   // 16KB A tile
  __shared__ int   toks[16];
  const int tid = threadIdx.x;
  const int m0  = blockIdx.x * 16;

  if (tid < 16) {
    int r = m0 + tid, b = r / STEPS, t = r % STEPS;
    toks[tid] = (t == 0) ? SOS : target[b * STEPS + t - 1];
  }
  __syncthreads();
  for (int idx = tid; idx < 16 * EMB; idx += 256)
    xA[idx] = embedding[(size_t)toks[idx >> 8] * EMB + (idx & 255)];
  __syncthreads();

  const int lane = tid & 31, wave = tid >> 5;
  const int l16 = lane & 15, half = lane >> 4;
  const float* aBase = xA + l16 * EMB + 2 * half;

  for (int q = 0; q < 4; ++q) {
    const int j0 = (wave + 8 * q) * 16;              // 0..496
    v8f a0 = {0.f,0.f,0.f,0.f,0.f,0.f,0.f,0.f};
    v8f a1 = a0, a2 = a0;
    const float* br = w_ih + (size_t)(j0 + l16) * EMB + 2 * half;
    const float* bz = w_ih + (size_t)(512 + j0 + l16) * EMB + 2 * half;
    const float* bn = w_ih + (size_t)(1024 + j0 + l16) * EMB + 2 * half;
#pragma unroll 4
    for (int k0 = 0; k0 < EMB; k0 += 4) {
      v2f a = *(const v2f*)(aBase + k0);
      a0 = wmma4(a, *(const v2f*)(br + k0), a0);
      a1 = wmma4(a, *(const v2f*)(bz + k0), a1);
      a2 = wmma4(a, *(const v2f*)(bn + k0), a2);
    }
    const int j = j0 + l16;
    const float cr = b_ih[j] + b_hh[j];
    const float cz = b_ih[512 + j] + b_hh[512 + j];
    const float cn = b_ih[1024 + j];
    const float bh = b_hh[1024 + j];
#pragma unroll
    for (int g = 0; g < 8; ++g) {
      const int i = g + 8 * half;
      float rg = 1.f / (1.f + __expf(-(a0[g] + cr)));
      float zg = 1.f / (1.f + __expf(-(a1[g] + cz)));
      float ng = tanhf(a2[g] + cn + rg * bh);
      float h  = (1.f - zg) * ng;
      const int r = m0 + i;
      hbuf[(size_t)r * HID + j] = h;
      if (r % STEPS == STEPS - 1) dhid[(size_t)(r / STEPS) * HID + j] = h;
    }
  }
}

// ============================================================================
// Kernel 2: dot-product attention. Block = (batch b, 4 timesteps).
// ============================================================================
#define TCH 4
#define TBLK 13   // ceil(51/4)
__global__ __launch_bounds__(256) void k2_attn(
    const float* __restrict__ hbuf, const float* __restrict__ enc,
    float* __restrict__ ctxbuf) {
  __shared__ float hb[TCH * HID];    // 8KB
  __shared__ float encC[16 * HID];   // 32KB
  __shared__ float sc[TCH * SRC];    // 2KB
  const int tid = threadIdx.x;
  const int b   = blockIdx.x / TBLK;
  const int t0  = (blockIdx.x % TBLK) * TCH;

  for (int idx = tid; idx < TCH * HID; idx += 256) {
    int tl = idx / HID, k = idx % HID, t = t0 + tl;
    hb[idx] = (t < STEPS) ? hbuf[(size_t)(b * STEPS + t) * HID + k] : 0.f;
  }
  __syncthreads();

  const int p = tid >> 2, sub = tid & 3;
  const int tl = p >> 4, sl = p & 15;
  for (int s0 = 0; s0 < SRC; s0 += 16) {
    for (int idx = tid; idx < 16 * HID / 4; idx += 256)
      ((v4f*)encC)[idx] =
          ((const v4f*)(enc + ((size_t)b * SRC + s0) * HID))[idx];
    __syncthreads();
    float dot = 0.f;
    const float* hr = hb + tl * HID + sub * 128;
    const float* er = encC + sl * HID + sub * 128;
#pragma unroll 4
    for (int k = 0; k < 128; ++k) dot += hr[k] * er[k];
    dot += __shfl_xor(dot, 1);
    dot += __shfl_xor(dot, 2);
    if (sub == 0) sc[tl * SRC + s0 + sl] = dot;
    __syncthreads();
  }
  if (tid < TCH) {
    float mx = -3.0e38f;
    for (int s = 0; s < SRC; ++s) mx = fmaxf(mx, sc[tid * SRC + s]);
    float sum = 0.f;
    for (int s = 0; s < SRC; ++s) {
      float e = __expf(sc[tid * SRC + s] - mx);
      sc[tid * SRC + s] = e;
      sum += e;
    }
    float inv = 1.f / sum;
    for (int s = 0; s < SRC; ++s) sc[tid * SRC + s] *= inv;
  }
  __syncthreads();
  float acc[TCH][2];
#pragma unroll
  for (int i = 0; i < TCH; ++i) { acc[i][0] = 0.f; acc[i][1] = 0.f; }
  const int c0 = tid, c1 = tid + 256;
  for (int s = 0; s < SRC; ++s) {
    float e0 = enc[((size_t)b * SRC + s) * HID + c0];
    float e1 = enc[((size_t)b * SRC + s) * HID + c1];
#pragma unroll
    for (int i = 0; i < TCH; ++i) {
      float w = sc[i * SRC + s];
      acc[i][0] += w * e0;
      acc[i][1] += w * e1;
    }
  }
#pragma unroll
  for (int i = 0; i < TCH; ++i) {
    int t = t0 + i;
    if (t < STEPS) {
      size_t r = (size_t)(b * STEPS + t);
      ctxbuf[r * HID + c0] = acc[i][0];
      ctxbuf[r * HID + c1] = acc[i][1];
    }
  }
}

// ============================================================================
// Kernel 3: o = tanh([h,ctx] @ wa^T), K=1024 as two 512 phases, 32KB LDS A.
// ============================================================================
__global__ __launch_bounds__(256) void k3_wa(
    const float* __restrict__ hbuf, const float* __restrict__ ctxbuf,
    const float* __restrict__ wa, float* __restrict__ obuf) {
  __shared__ float A[16 * HID];   // 32KB
  const int tid = threadIdx.x, m0 = blockIdx.x * 16;
  const int lane = tid & 31, wave = tid >> 5, l16 = lane & 15, half = lane >> 4;
  const float* aBase = A + l16 * HID + 2 * half;

  v8f acc[4];
#pragma unroll
  for (int q = 0; q < 4; ++q) acc[q] = (v8f){0.f,0.f,0.f,0.f,0.f,0.f,0.f,0.f};

  for (int idx = tid; idx < 16 * HID / 4; idx += 256)
    ((v4f*)A)[idx] = ((const v4f*)(hbuf + (size_t)m0 * HID))[idx];
  __syncthreads();
#pragma unroll
  for (int q = 0; q < 4; ++q) {
    const int n0 = (wave + 8 * q) * 16;
    const float* bp = wa + (size_t)(n0 + l16) * 1024 + 2 * half;
#pragma unroll 4
    for (int k0 = 0; k0 < HID; k0 += 4)
      acc[q] = wmma4(*(const v2f*)(aBase + k0), *(const v2f*)(bp + k0), acc[q]);
  }
  __syncthreads();
  for (int idx = tid; idx < 16 * HID / 4; idx += 256)
    ((v4f*)A)[idx] = ((const v4f*)(ctxbuf + (size_t)m0 * HID))[idx];
  __syncthreads();
#pragma unroll
  for (int q = 0; q < 4; ++q) {
    const int n0 = (wave + 8 * q) * 16;
    const float* bp = wa + (size_t)(n0 + l16) * 1024 + 512 + 2 * half;
#pragma unroll 4
    for (int k0 = 0; k0 < HID; k0 += 4)
      acc[q] = wmma4(*(const v2f*)(aBase + k0), *(const v2f*)(bp + k0), acc[q]);
#pragma unroll
    for (int g = 0; g < 8; ++g) {
      int row = m0 + g + 8 * half;
      obuf[(size_t)row * HID + n0 + l16] = tanhf(acc[q][g]);
    }
  }
}

// ============================================================================
// Kernel 4 (dominant, 107 GFLOP): logits = o @ fc_w^T + fc_b.
// Grid = (ROWS/32 row-groups) x (NSPLIT vocab slices) for wave-level
// parallelism; per-WG online softmax partials (m,s) -> pm/ps workspace.
// B fragments double-buffered in registers (8 k-steps per chunk) so the
// global loads clause up and the 16-WMMA burst hides their latency.
// Logit stores are non-temporal to keep fc_w (64MB) resident in L2.
// ============================================================================
#define K4R 32
#define NSPLIT 8
#define NT_PER_SPLIT (VOCAB / 16 / NSPLIT)   // 250
#define KCH 32                               // floats per k-chunk (8 k-steps)

__device__ __forceinline__ void k4_chunk(const float* aB0, const float* aB1,
                                         const v2f* bb, int kbase,
                                         v8f& acc0, v8f& acc1) {
#pragma unroll
  for (int i = 0; i < 8; ++i) {
    const int k0 = kbase + 4 * i;
    v2f a0 = *(const v2f*)(aB0 + k0);
    v2f a1 = *(const v2f*)(aB1 + k0);
    acc0 = wmma4(a0, bb[i], acc0);
    acc1 = wmma4(a1, bb[i], acc1);
  }
}

__global__ __launch_bounds__(256) void k4_logits(
    const float* __restrict__ obuf, const float* __restrict__ fc_w,
    const float* __restrict__ fc_b, float* __restrict__ out,
    float* __restrict__ pm, float* __restrict__ ps) {
  __shared__ float smem[K4R * HID];   // 64KB A tile, reused for stats reduce
  const int tid = threadIdx.x;
  const int mg  = blockIdx.x / NSPLIT;
  const int ns  = blockIdx.x % NSPLIT;
  const int m0  = mg * K4R;

  for (int idx = tid; idx < K4R * HID / 4; idx += 256)
    ((v4f*)smem)[idx] = ((const v4f*)(obuf + (size_t)m0 * HID))[idx];
  __syncthreads();

  const int lane = tid & 31, wave = tid >> 5, l16 = lane & 15, half = lane >> 4;
  const float* aB0 = smem + l16 * HID + 2 * half;
  const float* aB1 = smem + (16 + l16) * HID + 2 * half;

  float mt0[8], st0[8], mt1[8], st1[8];
#pragma unroll
  for (int g = 0; g < 8; ++g) {
    mt0[g] = -3.0e38f; st0[g] = 0.f;
    mt1[g] = -3.0e38f; st1[g] = 0.f;
  }

  const int ntEnd = (ns + 1) * NT_PER_SPLIT;
  for (int nt = ns * NT_PER_SPLIT + wave; nt < ntEnd; nt += 8) {
    const int n0 = nt * 16;
    const float* bp = fc_w + (size_t)(n0 + l16) * HID + 2 * half;
    v8f acc0 = {0.f,0.f,0.f,0.f,0.f,0.f,0.f,0.f};
    v8f acc1 = acc0;

    v2f bb0[8], bb1[8];
#pragma unroll
    for (int i = 0; i < 8; ++i) bb0[i] = *(const v2f*)(bp + 4 * i);
    // K = 512 -> 16 chunks of 32 floats; ping-pong bb0/bb1.
    for (int c = 0; c < 16; c += 2) {
#pragma unroll
      for (int i = 0; i < 8; ++i)
        bb1[i] = *(const v2f*)(bp + (c + 1) * KCH + 4 * i);
      k4_chunk(aB0, aB1, bb0, c * KCH, acc0, acc1);
      if (c + 2 < 16) {
#pragma unroll
        for (int i = 0; i < 8; ++i)
          bb0[i] = *(const v2f*)(bp + (c + 2) * KCH + 4 * i);
      }
      k4_chunk(aB0, aB1, bb1, (c + 1) * KCH, acc0, acc1);
    }

    const float bias = fc_b[n0 + l16];
#pragma unroll
    for (int g = 0; g < 8; ++g) {
      const int rr = g + 8 * half;
      float v0 = acc0[g] + bias;
      float v1 = acc1[g] + bias;
      __builtin_nontemporal_store(
          v0, &out[(size_t)(m0 + rr) * VOCAB + n0 + l16]);
      __builtin_nontemporal_store(
          v1, &out[(size_t)(m0 + 16 + rr) * VOCAB + n0 + l16]);
      float mn0 = fmaxf(mt0[g], v0);
      st0[g] = st0[g] * __expf(mt0[g] - mn0) + __expf(v0 - mn0);
      mt0[g] = mn0;
      float mn1 = fmaxf(mt1[g], v1);
      st1[g] = st1[g] * __expf(mt1[g] - mn1) + __expf(v1 - mn1);
      mt1[g] = mn1;
    }
  }

  // reduce (m,s) across the 16 lanes of each half-wave
#pragma unroll
  for (int off = 1; off < 16; off <<= 1) {
#pragma unroll
    for (int g = 0; g < 8; ++g) {
      float mo = __shfl_xor(mt0[g], off), so = __shfl_xor(st0[g], off);
      float mn = fmaxf(mt0[g], mo);
      st0[g] = st0[g] * __expf(mt0[g] - mn) + so * __expf(mo - mn);
      mt0[g] = mn;
      mo = __shfl_xor(mt1[g], off); so = __shfl_xor(st1[g], off);
      mn = fmaxf(mt1[g], mo);
      st1[g] = st1[g] * __expf(mt1[g] - mn) + so * __expf(mo - mn);
      mt1[g] = mn;
    }
  }
  __syncthreads();                 // A tile dead; reuse LDS for reduction
  float* red_m = smem;             // [8][K4R]
  float* red_s = smem + 8 * K4R;
  if (l16 == 0) {
#pragma unroll
    for (int g = 0; g < 8; ++g) {
      const int rr = g + 8 * half;
      red_m[wave * K4R + rr] = mt0[g];
      red_s[wave * K4R + rr] = st0[g];
      red_m[wave * K4R + 16 + rr] = mt1[g];
      red_s[wave * K4R + 16 + rr] = st1[g];
    }
  }
  __syncthreads();
  if (tid < K4R) {
    float m = red_m[tid], s = red_s[tid];
    for (int w = 1; w < 8; ++w) {
      float mo = red_m[w * K4R + tid], so = red_s[w * K4R + tid];
      float mn = fmaxf(m, mo);
      s = s * __expf(m - mn) + so * __expf(mo - mn);
      m = mn;
    }
    pm[(size_t)(mg * NSPLIT + ns) * K4R + tid] = m;
    ps[(size_t)(mg * NSPLIT + ns) * K4R + tid] = s;
  }
}

// ============================================================================
// Kernel 4b: fold the NSPLIT partial (m,s) per row into logZ.
// ============================================================================
__global__ __launch_bounds__(256) void k4b_logz(
    const float* __restrict__ pm, const float* __restrict__ ps,
    float* __restrict__ logZ) {
  int r = blockIdx.x * 256 + threadIdx.x;
  if (r >= ROWS) return;
  const int mg = r / K4R, row = r % K4R;
  float m = -3.0e38f, s = 0.f;
#pragma unroll
  for (int ns = 0; ns < NSPLIT; ++ns) {
    float mo = pm[(size_t)(mg * NSPLIT + ns) * K4R + row];
    float so = ps[(size_t)(mg * NSPLIT + ns) * K4R + row];
    float mn = fmaxf(m, mo);
    s = s * __expf(m - mn) + so * __expf(mo - mn);
    m = mn;
  }
  logZ[r] = m + __logf(s);
}

// ============================================================================
// Kernel 5: log_softmax fix-up: out -= logZ[row]  (non-temporal v4f stream)
// ============================================================================
__global__ __launch_bounds__(256) void k5_fix(float* __restrict__ out,
                                              const float* __restrict__ logZ) {
  size_t idx = (size_t)blockIdx.x * 256 + threadIdx.x;
  const size_t n4 = (size_t)ROWS * VOCAB / 4;
  if (idx >= n4) return;
  int r = (int)(idx / (VOCAB / 4));
  float z = logZ[r];
  v4f v = __builtin_nontemporal_load(((const v4f*)out) + idx);
  v.x -= z; v.y -= z; v.z -= z; v.w -= z;
  __builtin_nontemporal_store(v, ((v4f*)out) + idx);
}

// ============================================================================
extern "C" void kernel_launch(void* const* d_in, const int* in_sizes, int n_in,
                              void* d_out, int out_size, void* d_ws,
                              size_t ws_size, hipStream_t stream) {
  (void)in_sizes; (void)n_in; (void)out_size; (void)ws_size;
  const int*   target = (const int*)d_in[0];
  // d_in[1] = encoder_hidden (unused by the reference computation)
  const float* enc    = (const float*)d_in[2];
  const float* emb    = (const float*)d_in[3];
  const float* w_ih   = (const float*)d_in[4];
  // d_in[5] = w_hh (unused: h0 == 0)
  const float* b_ih   = (const float*)d_in[6];
  const float* b_hh   = (const float*)d_in[7];
  const float* wa     = (const float*)d_in[8];
  const float* fc_w   = (const float*)d_in[9];
  const float* fc_b   = (const float*)d_in[10];

  float* out  = (float*)d_out;
  float* dhid = out + (size_t)ROWS * VOCAB;         // decoder_hidden [1,64,512]

  float* hbuf   = (float*)d_ws;                     // [3264,512]
  float* ctxbuf = hbuf + (size_t)ROWS * HID;        // [3264,512]
  float* obuf   = ctxbuf + (size_t)ROWS * HID;      // [3264,512]
  float* logZ   = obuf + (size_t)ROWS * HID;        // [3264]
  float* pm     = logZ + ROWS;                      // [102*8*32]
  float* ps     = pm + (size_t)ROWS * NSPLIT;       // [102*8*32]

  hipLaunchKernelGGL(k1_gru, dim3(MT16), dim3(256), 0, stream,
                     target, emb, w_ih, b_ih, b_hh, hbuf, dhid);
  hipLaunchKernelGGL(k2_attn, dim3(BATCH * TBLK), dim3(256), 0, stream,
                     hbuf, enc, ctxbuf);
  hipLaunchKernelGGL(k3_wa, dim3(MT16), dim3(256), 0, stream,
                     hbuf, ctxbuf, wa, obuf);
  hipLaunchKernelGGL(k4_logits, dim3((ROWS / K4R) * NSPLIT), dim3(256), 0,
                     stream, obuf, fc_w, fc_b, out, pm, ps);
  hipLaunchKernelGGL(k4b_logz, dim3((ROWS + 255) / 256), dim3(256), 0, stream,
                     pm, ps, logZ);
  const unsigned nfix = (unsigned)(((size_t)ROWS * VOCAB / 4 + 255) / 256);
  hipLaunchKernelGGL(k5_fix, dim3(nfix), dim3(256), 0, stream, out, logZ);
}